// DisenGCNModel_9208409883325
// MI455X (gfx1250) — compile-verified
//
#include <hip/hip_runtime.h>
#include <hip/hip_bf16.h>
#include <math.h>

// ---------------------------------------------------------------------------
// DisenGCN forward on MI455X (gfx1250, wave32).
//
// Work: ~134 MFLOP GEMM + ~2 MB gathers + light routing -> latency bound.
// GEMMs run on the matrix pipe via V_WMMA_F32_16X16X4_F32 (full f32, no
// precision loss vs the f32 reference). A-fragment rows are CLAMPED (not
// zero-filled) at the ragged tail: in D = A*B, output row m depends only on
// A row m, and OOB output rows are never stored -- so clamping keeps control
// flow uniform and lets A load as one global_load_b64 per K-step instead of
// divergent exec-masked scalar loads.
// ---------------------------------------------------------------------------

typedef __attribute__((ext_vector_type(2))) float v2f;
typedef __attribute__((ext_vector_type(8))) float v8f;

#define EPS_NRM 1e-12f

// ---------------------------------------------------------------------------
// 1) Gather: eu = [Gu[user]; Gi[neigh_user]],  ei = [Gi[item]; Gu[neigh_item]]
//    One float4 per thread; blockIdx.y selects side.
// ---------------------------------------------------------------------------
__global__ void gather_kernel(const float* __restrict__ Gu,
                              const float* __restrict__ Gi,
                              const int* __restrict__ user,
                              const int* __restrict__ item,
                              const int* __restrict__ neigh_user,
                              const int* __restrict__ neigh_item,
                              float* __restrict__ eu,
                              float* __restrict__ ei,
                              int nneigh) {
  int idx = blockIdx.x * blockDim.x + threadIdx.x;  // over (nneigh+1)*16 float4s
  int total = (nneigh + 1) * 16;
  if (idx >= total) return;
  int row = idx >> 4;
  int seg = idx & 15;
  const float4* src;
  float4* dst;
  if (blockIdx.y == 0) {
    long g = (row == 0) ? (long)user[0] : (long)neigh_user[row - 1];
    src = (row == 0) ? (const float4*)(Gu + g * 64) : (const float4*)(Gi + g * 64);
    dst = (float4*)(eu + (long)row * 64);
  } else {
    long g = (row == 0) ? (long)item[0] : (long)neigh_item[row - 1];
    src = (row == 0) ? (const float4*)(Gi + g * 64) : (const float4*)(Gu + g * 64);
    dst = (float4*)(ei + (long)row * 64);
  }
  dst[seg] = src[seg];
}

// ---------------------------------------------------------------------------
// 2) z[n, k*16+d] = relu( sum_i e[n,i] * W[k,i,d] + b[k,d] )
//    W: (4,64,16) flat. Logical GEMM (nrows x 64) @ (64 x 64).
//    One wave -> 16 rows x 64 cols via 4 f32 WMMA accumulators, K stepped by 4.
//    A(16x4) layout: lanes 0-15 hold (M=lane, K=0,1), lanes 16-31 (M=lane-16, K=2,3).
//    B(4x16) layout mirrored. C/D: vgpr v -> rows v / v+8 for lane halves.
// ---------------------------------------------------------------------------
__global__ __launch_bounds__(128)
void gemm_bias_relu(const float* __restrict__ eU, const float* __restrict__ eI,
                    float* __restrict__ zU, float* __restrict__ zI,
                    const float* __restrict__ W, const float* __restrict__ b,
                    int nrows) {
  const float* e = (blockIdx.y == 0) ? eU : eI;
  float* z = (blockIdx.y == 0) ? zU : zI;

  int wave = blockIdx.x * (blockDim.x >> 5) + (threadIdx.x >> 5);
  int lane = threadIdx.x & 31;
  int row0 = wave * 16;
  if (row0 >= nrows) return;  // uniform per wave: EXEC stays all-ones for WMMA

  int lo = lane & 15;
  int koff = (lane >= 16) ? 2 : 0;  // lanes 16-31 carry K+2 / K+3
  int mrow = row0 + lo;             // A-fragment row
  int mrowc = mrow < nrows ? mrow : (nrows - 1);  // clamp ragged tail (see header)

  // Hoisted per-lane bases: all K-step loads become immediate-offset accesses.
  const float* __restrict__ ea = e + (long)mrowc * 64 + koff;  // A: 2 floats, 8B aligned
  const float* __restrict__ wb = W + koff * 16 + lo;           // B: Wf(i,col)=W[t*1024+i*16+lo]

  v8f acc0 = {}, acc1 = {}, acc2 = {}, acc3 = {};

#pragma unroll
  for (int k = 0; k < 64; k += 4) {
    v2f a = *(const v2f*)(ea + k);  // (K=k+koff, K=k+koff+1) for this lane's row

#define B_FRAG(t)                                      \
    ({ v2f bb;                                         \
       bb.x = wb[(t) * 1024 + k * 16];                 \
       bb.y = wb[(t) * 1024 + k * 16 + 16];            \
       bb; })

    acc0 = __builtin_amdgcn_wmma_f32_16x16x4_f32(false, a, false, B_FRAG(0),
                                                 (short)0, acc0, false, false);
    acc1 = __builtin_amdgcn_wmma_f32_16x16x4_f32(false, a, false, B_FRAG(1),
                                                 (short)0, acc1, false, false);
    acc2 = __builtin_amdgcn_wmma_f32_16x16x4_f32(false, a, false, B_FRAG(2),
                                                 (short)0, acc2, false, false);
    acc3 = __builtin_amdgcn_wmma_f32_16x16x4_f32(false, a, false, B_FRAG(3),
                                                 (short)0, acc3, false, false);
#undef B_FRAG
  }

  // Epilogue: bias + ReLU + store. Element (t, v): row = row0+v+(hi?8:0), col = t*16+lo.
  int rbase = row0 + ((lane >= 16) ? 8 : 0);
  v8f accs[4] = {acc0, acc1, acc2, acc3};
#pragma unroll
  for (int t = 0; t < 4; ++t) {
    int col = t * 16 + lo;
    float bias = b[col];
#pragma unroll
    for (int v = 0; v < 8; ++v) {
      int r = rbase + v;
      if (r < nrows) {
        float val = accs[t][v] + bias;
        z[(long)r * 64 + col] = val > 0.0f ? val : 0.0f;
      }
    }
  }
}

// ---------------------------------------------------------------------------
// 3) Normalize neighbor capsules once per layer: zn[n,k,:] = z[n,k,:]/max(|.|,eps)
//    rows 1..nneigh, one (row,k) capsule per thread.
// ---------------------------------------------------------------------------
__global__ void normalize_nb(const float* __restrict__ zU, const float* __restrict__ zI,
                             float* __restrict__ znU, float* __restrict__ znI,
                             int nneigh) {
  int idx = blockIdx.x * blockDim.x + threadIdx.x;  // over nneigh*4 capsules
  if (idx >= nneigh * 4) return;
  int row = 1 + (idx >> 2);
  int k = idx & 3;
  const float* z = (blockIdx.y == 0) ? zU : zI;
  float* zn = (blockIdx.y == 0) ? znU : znI;
  const float* p = z + (long)row * 64 + k * 16;
  float* q = zn + (long)row * 64 + k * 16;
  float v[16], s = 0.0f;
#pragma unroll
  for (int d = 0; d < 16; ++d) { v[d] = p[d]; s += v[d] * v[d]; }
  float nrm = sqrtf(s);
  nrm = nrm > EPS_NRM ? nrm : EPS_NRM;
  float inv = 1.0f / nrm;
#pragma unroll
  for (int d = 0; d < 16; ++d) q[d] = v[d] * inv;
}

// ---------------------------------------------------------------------------
// 4) Routing: 3 iterations of { normalize ego; p = softmax_k(zn . ego);
//    ego = sum_n p[n,k]*zn[n,k,:] }. One workgroup per side (blockIdx.x).
//    n-reduction: per-lane accumulate -> wave32 shfl_xor reduce -> LDS atomics.
// ---------------------------------------------------------------------------
__global__ __launch_bounds__(256)
void routing_kernel(float* __restrict__ zU, float* __restrict__ zI,
                    const float* __restrict__ znU, const float* __restrict__ znI,
                    int nneigh) {
  float* z = (blockIdx.x == 0) ? zU : zI;
  const float* zn = (blockIdx.x == 0) ? znU : znI;

  __shared__ float ego[64];     // raw ego z[0]
  __shared__ float egon[64];    // normalized ego
  __shared__ float nrmk[4];
  __shared__ float newego[64];

  int tid = threadIdx.x;
  if (tid < 64) ego[tid] = z[tid];
  __syncthreads();

  for (int it = 0; it < 3; ++it) {
    if (tid < 4) {
      float s = 0.0f;
#pragma unroll
      for (int d = 0; d < 16; ++d) { float v = ego[tid * 16 + d]; s += v * v; }
      float nr = sqrtf(s);
      nrmk[tid] = nr > EPS_NRM ? nr : EPS_NRM;
    }
    __syncthreads();
    if (tid < 64) {
      egon[tid] = ego[tid] / nrmk[tid >> 4];
      newego[tid] = 0.0f;
    }
    __syncthreads();

    float acc[64];
#pragma unroll
    for (int j = 0; j < 64; ++j) acc[j] = 0.0f;

    for (int n = 1 + tid; n <= nneigh; n += 256) {
      const float* r = zn + (long)n * 64;
      float dot[4] = {0.f, 0.f, 0.f, 0.f};
#pragma unroll
      for (int k = 0; k < 4; ++k)
#pragma unroll
        for (int d = 0; d < 16; ++d)
          dot[k] += r[k * 16 + d] * egon[k * 16 + d];
      float m = fmaxf(fmaxf(dot[0], dot[1]), fmaxf(dot[2], dot[3]));
      float ex[4], s = 0.0f;
#pragma unroll
      for (int k = 0; k < 4; ++k) { ex[k] = expf(dot[k] - m); s += ex[k]; }
      float inv = 1.0f / s;
#pragma unroll
      for (int k = 0; k < 4; ++k) {
        float p = ex[k] * inv;
#pragma unroll
        for (int d = 0; d < 16; ++d)
          acc[k * 16 + d] += p * r[k * 16 + d];  // r re-read: L0/L2 hot, saves VGPRs
      }
    }

    // wave32 reduce each of 64 partials, then one LDS atomic per wave
#pragma unroll
    for (int j = 0; j < 64; ++j) {
      float v = acc[j];
#pragma unroll
      for (int off = 16; off > 0; off >>= 1)
        v += __shfl_xor(v, off, 32);
      if ((tid & 31) == 0) atomicAdd(&newego[j], v);
    }
    __syncthreads();
    if (tid < 64) ego[tid] = newego[tid];
    __syncthreads();
  }
  if (tid < 64) z[tid] = ego[tid];
}

// ---------------------------------------------------------------------------
// 5) Finalize: out[0] = dot(eu0, ei0); out[1..64] = eu0; out[65..128] = ei0.
// ---------------------------------------------------------------------------
__global__ void finalize_kernel(const float* __restrict__ zU,
                                const float* __restrict__ zI,
                                float* __restrict__ out) {
  __shared__ float red[64];
  int tid = threadIdx.x;  // 64 threads
  float u = zU[tid];
  float i = zI[tid];
  red[tid] = u * i;
  out[1 + tid] = u;
  out[65 + tid] = i;
  __syncthreads();
  if (tid == 0) {
    float s = 0.0f;
    for (int j = 0; j < 64; ++j) s += red[j];
    out[0] = s;
  }
}

// ---------------------------------------------------------------------------
// Host-side launch
// ---------------------------------------------------------------------------
extern "C" void kernel_launch(void* const* d_in, const int* in_sizes, int n_in,
                              void* d_out, int out_size, void* d_ws, size_t ws_size,
                              hipStream_t stream) {
  const int* user = (const int*)d_in[0];
  const int* item = (const int*)d_in[1];
  const int* neigh_user = (const int*)d_in[2];
  const int* neigh_item = (const int*)d_in[3];
  const float* Gu = (const float*)d_in[4];
  const float* Gi = (const float*)d_in[5];
  const float* W0 = (const float*)d_in[6];
  const float* b0 = (const float*)d_in[7];
  const float* W1 = (const float*)d_in[8];
  const float* b1 = (const float*)d_in[9];
  float* out = (float*)d_out;

  const int nneigh = in_sizes[2];   // 4096
  const int N = nneigh + 1;         // 4097 rows

  float* ws = (float*)d_ws;
  const long stride = (long)N * 64;
  float* eu  = ws;
  float* ei  = eu  + stride;
  float* zu  = ei  + stride;
  float* zi  = zu  + stride;
  float* znu = zi  + stride;
  float* zni = znu + stride;

  // 1) gather
  {
    int total = N * 16;
    dim3 grid((total + 255) / 256, 2);
    gather_kernel<<<grid, 256, 0, stream>>>(Gu, Gi, user, item,
                                            neigh_user, neigh_item, eu, ei, nneigh);
  }

  const int waves = (N + 15) / 16;                 // 257 row tiles
  dim3 gemm_grid((waves + 3) / 4, 2);              // 4 waves / 128-thread block
  dim3 norm_grid((nneigh * 4 + 255) / 256, 2);

  // 2) layer 0
  gemm_bias_relu<<<gemm_grid, 128, 0, stream>>>(eu, ei, zu, zi, W0, b0, N);
  normalize_nb<<<norm_grid, 256, 0, stream>>>(zu, zi, znu, zni, nneigh);
  routing_kernel<<<2, 256, 0, stream>>>(zu, zi, znu, zni, nneigh);

  // 3) layer 1 (input = routed zu/zi, output into eu/ei)
  gemm_bias_relu<<<gemm_grid, 128, 0, stream>>>(zu, zi, eu, ei, W1, b1, N);
  normalize_nb<<<norm_grid, 256, 0, stream>>>(eu, ei, znu, zni, nneigh);
  routing_kernel<<<2, 256, 0, stream>>>(eu, ei, znu, zni, nneigh);

  // 4) outputs
  finalize_kernel<<<1, 64, 0, stream>>>(eu, ei, out);
}